// BaseLinearOdeLayer_70643622085074
// MI455X (gfx1250) — compile-verified
//
#include <hip/hip_runtime.h>
#include <hip/hip_bf16.h>

typedef __attribute__((ext_vector_type(2))) float v2f;
typedef __attribute__((ext_vector_type(8))) float v8f;

#define NSAMP  256
#define DIM    64
#define NT     64
#define JTERMS 32            // Taylor order == K of the combine GEMM
#define LDA    65            // padded LDS stride (bank-conflict-free transpose)

__global__ __launch_bounds__(256)
void linode_krylov_wmma(const float* __restrict__ A,
                        const float* __restrict__ x_init,
                        const float* __restrict__ t,
                        float* __restrict__ out)
{
    __shared__ float As[DIM * LDA];      // As[k*LDA + r] = A[r][k]  (transposed)
    __shared__ float V[JTERMS * DIM];    // V[j*64 + d] = (A^j x)_d / j!
    __shared__ float ts[NT];

    const int n   = blockIdx.x;
    const int tid = threadIdx.x;

    const float*  An  = A + (size_t)n * DIM * DIM;
    const float4* An4 = (const float4*)An;           // 1024 float4 per sample

    // ---- Phase 1: coalesced load of A, transposed into LDS ----
    #pragma unroll
    for (int s = 0; s < 4; ++s) {
        int i4 = tid + 256 * s;                      // float4 index, [0,1024)
        float4 v = An4[i4];
        int r  = i4 >> 4;                            // row    = (i4*4)/64
        int k0 = (i4 & 15) << 2;                     // column base
        As[(k0 + 0) * LDA + r] = v.x;
        As[(k0 + 1) * LDA + r] = v.y;
        As[(k0 + 2) * LDA + r] = v.z;
        As[(k0 + 3) * LDA + r] = v.w;
    }
    if (tid < NT)  ts[tid] = t[n * NT + tid];
    if (tid < DIM) V[tid]  = x_init[n * DIM + tid];  // v_0 = x
    __syncthreads();

    // ---- Phase 2: Krylov chain  v_j = (A v_{j-1}) / j  (factorial folded in) ----
    for (int j = 1; j < JTERMS; ++j) {
        if (tid < DIM) {
            float acc = 0.f;
            #pragma unroll 8
            for (int k = 0; k < DIM; ++k)
                acc += As[k * LDA + tid] * V[(j - 1) * DIM + k];
            V[j * DIM + tid] = acc * (1.0f / (float)j);
        }
        __syncthreads();
    }

    // ---- Phase 3: out[n][kt][d] = sum_j ts[kt]^j * V[j][d]
    //      64x64x32 GEMM per sample via V_WMMA_F32_16X16X4_F32 (fp32 matrix core)
    const int wave = tid >> 5;
    const int lane = tid & 31;
    const int hi   = lane >> 4;        // lane half selects K-pair / M-offset
    const int lo   = lane & 15;

    float* outn = out + (size_t)n * NT * DIM;

    #pragma unroll
    for (int rep = 0; rep < 2; ++rep) {
        const int tile = wave + 8 * rep;             // 0..15
        const int tm   = tile >> 2;                  // 16-row time block
        const int tn   = tile & 3;                   // 16-col dim block

        // A-fragment generator: lane M = lo, needs t^(j0 + 2*hi + {0,1})
        const float tv = ts[tm * 16 + lo];
        const float t2 = tv * tv;
        const float t4 = t2 * t2;
        float p = hi ? t2 : 1.0f;                    // t^(2*hi)

        v8f c = {};
        #pragma unroll
        for (int kk = 0; kk < JTERMS / 4; ++kk) {
            const int j0 = kk * 4;
            v2f a, b;
            a.x = p;                                  // T[M][j0+2hi]
            a.y = p * tv;                             // T[M][j0+2hi+1]
            b.x = V[(j0 + 2 * hi)     * DIM + tn * 16 + lo];  // K rows {0,2}+j0
            b.y = V[(j0 + 2 * hi + 1) * DIM + tn * 16 + lo];  // K rows {1,3}+j0
            // 8 args: (neg_a, A, neg_b, B, c_mod, C, reuse_a, reuse_b)
            c = __builtin_amdgcn_wmma_f32_16x16x4_f32(false, a, false, b,
                                                      (short)0, c, false, false);
            p *= t4;                                  // advance running power by t^4
        }

        // C/D layout: VGPR i -> M = i + 8*hi, N = lo
        #pragma unroll
        for (int i = 0; i < 8; ++i) {
            int row = tm * 16 + i + 8 * hi;          // time index
            int col = tn * 16 + lo;                  // dim index
            outn[row * DIM + col] = c[i];
        }
    }
}

extern "C" void kernel_launch(void* const* d_in, const int* in_sizes, int n_in,
                              void* d_out, int out_size, void* d_ws, size_t ws_size,
                              hipStream_t stream)
{
    const float* A = (const float*)d_in[0];
    const float* x = (const float*)d_in[1];
    const float* t = (const float*)d_in[2];
    float* out = (float*)d_out;
    (void)in_sizes; (void)n_in; (void)out_size; (void)d_ws; (void)ws_size;

    linode_krylov_wmma<<<dim3(NSAMP), dim3(256), 0, stream>>>(A, x, t, out);
}